// BatchGeneralization_70111046139941
// MI455X (gfx1250) — compile-verified
//
#include <hip/hip_runtime.h>
#include <stdint.h>

typedef float float4_t __attribute__((ext_vector_type(4)));
typedef int   v4i      __attribute__((vector_size(16)));   // matches builtin param

#define AS1 __attribute__((address_space(1)))
#define AS3 __attribute__((address_space(3)))

#if __has_builtin(__builtin_amdgcn_global_load_async_to_lds_b128) && \
    __has_builtin(__builtin_amdgcn_global_store_async_from_lds_b128) && \
    __has_builtin(__builtin_amdgcn_s_wait_asynccnt)
#define USE_ASYNC_LDS 1
#else
#define USE_ASYNC_LDS 0
#endif

#define COPY_THREADS 256
#define CHUNK_BYTES (COPY_THREADS * 16)   // 4 KiB per chunk: 16B per lane

// ---------------------------------------------------------------------------
// Kernel 1: bulk copy x -> out, streamed through LDS via gfx1250 async copies
// (GLOBAL_LOAD_ASYNC_TO_LDS_B128 / GLOBAL_STORE_ASYNC_FROM_LDS_B128, tracked
// by ASYNCcnt). Per-WGP HBM share is far below LDS bandwidth, so LDS staging
// is free; async ops keep VGPR pressure near zero. Wave-level parallelism
// (8 wave32 per block x 2048 blocks) hides the per-wave load->store wait.
// ---------------------------------------------------------------------------
__global__ __launch_bounds__(COPY_THREADS)
void bg_bulk_copy(const float* __restrict__ x, float* __restrict__ out,
                  long nChunks, long nFloats)
{
#if USE_ASYNC_LDS
    __shared__ __align__(16) char stage[CHUNK_BYTES];
    const unsigned laneOff = threadIdx.x * 16u;
    AS3 v4i* ldsSlot = (AS3 v4i*)(unsigned)(uintptr_t)(&stage[laneOff]);

    for (long c = blockIdx.x; c < nChunks; c += gridDim.x) {
        uintptr_t g = (uintptr_t)x   + (uintptr_t)c * CHUNK_BYTES + laneOff;
        uintptr_t o = (uintptr_t)out + (uintptr_t)c * CHUNK_BYTES + laneOff;
        __builtin_amdgcn_global_load_async_to_lds_b128((AS1 v4i*)g, ldsSlot, 0, 0);
        __builtin_amdgcn_s_wait_asynccnt(0);   // load done; also retires prior store
        __builtin_amdgcn_global_store_async_from_lds_b128((AS1 v4i*)o, ldsSlot, 0, 0);
    }
    __builtin_amdgcn_s_wait_asynccnt(0);
#else
    // Fallback: plain B128 streaming copy.
    const long n4 = nChunks * (CHUNK_BYTES / 16);
    const float4_t* __restrict__ s4 = (const float4_t*)x;
    float4_t* __restrict__ d4 = (float4_t*)out;
    for (long i = (long)blockIdx.x * blockDim.x + threadIdx.x; i < n4;
         i += (long)gridDim.x * blockDim.x) {
        d4[i] = s4[i];
    }
#endif
    // Scalar tail (empty for the reference shapes, kept for generality).
    const long tail = nChunks * (CHUNK_BYTES / 4);
    for (long i = tail + (long)blockIdx.x * blockDim.x + threadIdx.x; i < nFloats;
         i += (long)gridDim.x * blockDim.x) {
        out[i] = x[i];
    }
}

// ---------------------------------------------------------------------------
// Kernel 2: overwrite the 819 ref rows with m*x[tgt] + (1-m)*x[ref].
// One block per mixed row; float4 vectorized; reads ORIGINAL x (never out),
// so ordering within the row set does not matter (ref_index is unique).
// ---------------------------------------------------------------------------
__global__ __launch_bounds__(256)
void bg_mix_rows(const float* __restrict__ x,
                 const int* __restrict__ refIdx,
                 const int* __restrict__ tgtIdx,
                 const float* __restrict__ mag,
                 float* __restrict__ out, int D)
{
    const int i = blockIdx.x;
    const long r = (long)refIdx[i];
    const long t = (long)tgtIdx[i];
    const float m  = mag[i];
    const float om = 1.0f - m;

    const float4_t* __restrict__ xr = (const float4_t*)(x + r * (long)D);
    const float4_t* __restrict__ xt = (const float4_t*)(x + t * (long)D);
    float4_t* __restrict__ o = (float4_t*)(out + r * (long)D);

    const int n4 = D >> 2;
    for (int j = threadIdx.x; j < n4; j += blockDim.x) {
        __builtin_prefetch(&xt[j + blockDim.x], 0, 0);  // global_prefetch_b8
        __builtin_prefetch(&xr[j + blockDim.x], 0, 0);
        float4_t a = xt[j];
        float4_t b = xr[j];
        o[j] = a * m + b * om;   // matches reference: x[t]*m + x[r]*(1-m)
    }
}

// ---------------------------------------------------------------------------
// Harness entry. Inputs (setup_inputs order): x, y, ref_index, target_index,
// mag. y is unused by the computation.
// ---------------------------------------------------------------------------
extern "C" void kernel_launch(void* const* d_in, const int* in_sizes, int n_in,
                              void* d_out, int out_size, void* d_ws, size_t ws_size,
                              hipStream_t stream)
{
    const float* x    = (const float*)d_in[0];
    const int*   ref  = (const int*)  d_in[2];
    const int*   tgt  = (const int*)  d_in[3];
    const float* mag  = (const float*)d_in[4];
    float*       out  = (float*)d_out;

    const long nFloats = (long)in_sizes[0];              // B*D = 33,554,432
    const int  B       = in_sizes[1];                    // 8192
    const int  D       = (int)(nFloats / B);             // 4096
    const int  nMix    = in_sizes[2];                    // 819

    const long nChunks = (nFloats * 4) / CHUNK_BYTES;    // full 4 KiB chunks

    // Bulk copy: 2048 blocks x 256 threads (8 wave32 per block) grid-striding
    // over 32768 chunks — enough waves in flight to saturate 23.3 TB/s.
    bg_bulk_copy<<<2048, COPY_THREADS, 0, stream>>>(x, out, nChunks, nFloats);

    // Row blend: one block per mixed row; runs after the copy on the same
    // stream, so its writes win on the 819 ref rows.
    if (nMix > 0) {
        bg_mix_rows<<<nMix, 256, 0, stream>>>(x, ref, tgt, mag, out, D);
    }
}